// DifferentiableExtrusionFast_6622839570470
// MI455X (gfx1250) — compile-verified
//
#include <hip/hip_runtime.h>
#include <hip/hip_bf16.h>
#include <math.h>

typedef float v2f __attribute__((ext_vector_type(2)));
typedef float v8f __attribute__((ext_vector_type(8)));

#define SHARP 100.0f
#define EPSF  1e-8f

// polygons: (8,32,32,2) f32, attributes: (8,4) f32, validity: (8,32) f32
// out: (8,64,64,64) f32  laid out [b][z][y][x]
__global__ __launch_bounds__(256) void extrusion_wmma_kernel(
    const float* __restrict__ polygons,
    const float* __restrict__ attributes,
    const float* __restrict__ validity,
    float* __restrict__ out)
{
    __shared__ float2 poly[1024];   // 32 polygons x 32 vertices (x,y)
    __shared__ float2 aux[1024];    // per edge: {sx, 1/(|e|^2+eps)}
    __shared__ float  validv[32];

    const int tid    = threadIdx.x;
    const int lane   = tid & 31;
    const int nsub   = lane & 15;
    const bool hi    = lane >= 16;
    const int bitoff = hi ? 8 : 0;
    const int wave   = tid >> 5;
    const int b      = blockIdx.x >> 5;   // 8 batches
    const int blkInB = blockIdx.x & 31;   // 32 blocks per batch

    // ---- stage this batch's polygons (8KB) + validity into LDS ----
    const float2* gp = (const float2*)(polygons + (size_t)b * 2048);
    for (int i = tid; i < 1024; i += 256) poly[i] = gp[i];
    if (tid < 32) validv[tid] = validity[b * 32 + tid];
    __syncthreads();

    // ---- precompute division-heavy per-edge terms ONCE per block ----
    for (int i = tid; i < 1024; i += 256) {
        int p = i & 31;
        float2 v0 = poly[i];
        float2 v1 = poly[(i & ~31) | ((p + 1) & 31)];
        float ex = v1.x - v0.x, ey = v1.y - v0.y;
        float e2 = ex * ex + ey * ey + EPSF;
        float2 a;
        a.x = (v1.x - v0.x) / (v1.y - v0.y + EPSF);  // sx
        a.y = 1.0f / e2;                             // re2
        aux[i] = a;
    }
    __syncthreads();

    // ---- this wave's 16 pixels: one row y, 16 consecutive x ----
    const int seg   = blkInB * 8 + wave;   // 0..255 segments per batch
    const int yrow  = seg >> 2;
    const int xbase = (seg & 3) << 4;

    const float inv = 1.0f / 63.0f;
    const float yp  = (float)yrow * inv;               // wave-uniform
    const float px  = (float)(xbase + nsub) * inv;     // per pixel N=lane%16

    // B operand (pixels on N), K = [px, py, 1, px^2+py^2]
    v2f Bpix;
    Bpix.x = hi ? 1.0f : px;
    Bpix.y = hi ? (px * px + yp * yp) : yp;

    float comb = 0.0f;   // per-lane result: pixel xbase + lane%16

    for (int n = 0; n < 32; ++n) {
        float mind = 3.0e38f;
        float cnt  = 0.0f;

        for (int pb = 0; pb < 32; pb += 16) {
            const int p  = pb + nsub;                  // edge this lane serves
            const int ei = n * 32 + p;
            float2 v0 = poly[ei];
            float2 v1 = poly[n * 32 + ((p + 1) & 31)];
            float2 ax = aux[ei];
            const float sx  = ax.x;
            const float re2 = ax.y;

            float ex = v1.x - v0.x;
            float ey = v1.y - v0.y;
            float e2 = ex * ex + ey * ey + EPSF;
            float cdot = -(v0.x * ex + v0.y * ey);
            float cvv  = v0.x * v0.x + v0.y * v0.y;
            float ccx  = v0.x - sx * v0.y;

            // crossing y-interval test: one bit per edge via ballot
            bool ycl = (v0.y <= yp) != (v1.y <= yp);
            unsigned long long cmask = __ballot(ycl);

            // A operands (edges on M): lanes0-15 hold K0/K1, lanes16-31 K2/K3
            v2f Adot, Atu, Avv, Acx, Ae2;
            Adot.x = hi ? cdot        : ex;
            Adot.y = hi ? 0.0f        : ey;
            Atu.x  = hi ? cdot * re2  : ex * re2;     // D = (v.e)/|e|^2
            Atu.y  = hi ? 0.0f        : ey * re2;
            Avv.x  = hi ? cvv         : -2.0f * v0.x; // D = |p-v0|^2
            Avv.y  = hi ? 1.0f        : -2.0f * v0.y;
            Acx.x  = hi ? ccx         : -1.0f;        // D = inter_x - px
            Acx.y  = hi ? 0.0f        : sx;
            Ae2.x  = hi ? e2          : 0.0f;         // D = e2[M] broadcast
            Ae2.y  = 0.0f;

            v8f zc = {};
            v8f Ddot = __builtin_amdgcn_wmma_f32_16x16x4_f32(false, Adot, false, Bpix, (short)0, zc, false, false);
            v8f Dtu  = __builtin_amdgcn_wmma_f32_16x16x4_f32(false, Atu,  false, Bpix, (short)0, zc, false, false);
            v8f Dvv  = __builtin_amdgcn_wmma_f32_16x16x4_f32(false, Avv,  false, Bpix, (short)0, zc, false, false);
            v8f Dcx  = __builtin_amdgcn_wmma_f32_16x16x4_f32(false, Acx,  false, Bpix, (short)0, zc, false, false);
            v8f De2  = __builtin_amdgcn_wmma_f32_16x16x4_f32(false, Ae2,  false, Bpix, (short)0, zc, false, false);

            #pragma unroll
            for (int r = 0; r < 8; ++r) {
                // element (edge M = r + 8*hi, pixel N = lane%16)
                float t  = fminf(fmaxf(Dtu[r], 0.0f), 1.0f);
                float d2 = Dvv[r] - t * (2.0f * Ddot[r] - t * De2[r]);
                mind = fminf(mind, sqrtf(fmaxf(d2, 0.0f)));
                bool yc = (cmask >> (r + bitoff)) & 1ull;
                cnt += (yc && (Dcx[r] > 0.0f)) ? 1.0f : 0.0f;
            }
        }

        // merge lane halves: edges r (lo lanes) vs r+8 (hi lanes)
        mind = fminf(mind, __shfl_xor(mind, 16, 32));
        cnt += __shfl_xor(cnt, 16, 32);

        float sdf   = ((((int)(cnt + 0.5f)) & 1) != 0) ? -mind : mind;
        bool  valid = validv[n] >= 0.5f;
        float m     = valid ? (1.0f / (1.0f + __expf(SHARP * sdf))) : 0.0f;
        comb = fmaxf(comb, m);
    }

    // ---- z-extrusion: h = clip(floor(attr0*64), 1, 64) ----
    float hf = attributes[b * 4 + 0];
    int h = (int)floorf(hf * 64.0f);
    h = min(max(h, 1), 64);

    const int x    = xbase + nsub;
    const int zoff = hi ? 32 : 0;               // lane halves split z range
    float* obase = out + (size_t)b * 262144 + (size_t)yrow * 64 + x;
    #pragma unroll 4
    for (int z2 = 0; z2 < 32; ++z2) {
        int zz = zoff + z2;
        obase[(size_t)zz * 4096] = (zz < h) ? comb : 0.0f;
    }
}

extern "C" void kernel_launch(void* const* d_in, const int* in_sizes, int n_in,
                              void* d_out, int out_size, void* d_ws, size_t ws_size,
                              hipStream_t stream) {
    (void)in_sizes; (void)n_in; (void)out_size; (void)d_ws; (void)ws_size;
    const float* polygons   = (const float*)d_in[0];  // 8*32*32*2
    const float* attributes = (const float*)d_in[1];  // 8*4
    const float* validity   = (const float*)d_in[2];  // 8*32
    float* out = (float*)d_out;                       // 8*64*64*64

    // 8 batches * 32 blocks; 256 threads = 8 wave32 waves per block
    extrusion_wmma_kernel<<<dim3(256), dim3(256), 0, stream>>>(
        polygons, attributes, validity, out);
}